// EfficientDet_50818053047095
// MI455X (gfx1250) — compile-verified
//
#include <hip/hip_runtime.h>
#include <hip/hip_bf16.h>
#include <math.h>
#include <stdint.h>

// ---------------------------------------------------------------------------
// EfficientDet post-process for MI455X (gfx1250, wave32).
//   Phase 1: per-anchor class max/argmax + box decode (streaming, ~133MB ->
//            HBM-bound, ~6us roofline at 23.3 TB/s)
//   Phase 2: greedy NMS, 1 workgroup / image; per-image score vector lives in
//            LDS (196KB of the 320KB WGP LDS). Preload uses BOTH CDNA5 bulk
//            copy engines: TENSOR_LOAD_TO_LDS (TDM, TENSORcnt) for the first
//            half and GLOBAL_LOAD_ASYNC_TO_LDS_B128 (ASYNCcnt) for the rest.
//   Phase 3: gather + re-decode kept boxes.
// No WMMA: the workload contains zero matrix products (reductions, decode,
// sequential greedy suppression), so matrix cores cannot be applied.
// ---------------------------------------------------------------------------

#define NCLS      80
#define MAXD      100
#define SCORE_TH  0.2f
#define IOU_TH    0.2f
#define NMS_THREADS 1024

#if defined(__has_builtin)
#if __has_builtin(__builtin_amdgcn_tensor_load_to_lds)
#define HAVE_TDM 1
#endif
#endif

typedef unsigned int u32x4 __attribute__((ext_vector_type(4)));
typedef int          i32x4 __attribute__((ext_vector_type(4)));
typedef int          i32x8 __attribute__((ext_vector_type(8)));

__device__ __forceinline__ float4 decode_box(const float4 an, const float4 lc, float size) {
    // an = (y1,x1,y2,x2) anchor ; lc = (dy,dx,dh,dw)
    float ya = (an.x + an.z) * 0.5f;
    float xa = (an.y + an.w) * 0.5f;
    float ha = an.z - an.x;
    float wa = an.w - an.y;
    float h  = expf(lc.z) * ha;
    float w  = expf(lc.w) * wa;
    float yc = lc.x * ha + ya;
    float xc = lc.y * wa + xa;
    float lim = size - 1.0f;
    float x1 = fminf(fmaxf(xc - w * 0.5f, 0.0f), lim);
    float y1 = fminf(fmaxf(yc - h * 0.5f, 0.0f), lim);
    float x2 = fminf(fmaxf(xc + w * 0.5f, 0.0f), lim);
    float y2 = fminf(fmaxf(yc + h * 0.5f, 0.0f), lim);
    return make_float4(x1, y1, x2, y2);   // (x1,y1,x2,y2)
}

// ---------------------------------------------------------------------------
// Phase 1: score = max_c cls[b,a,c]; label = argmax (first index on ties);
// decode box, apply class offset (label * 2*size), thresholded score.
// ---------------------------------------------------------------------------
__global__ __launch_bounds__(256, 1)
void det_prep(const float* __restrict__ cls, const float* __restrict__ loc,
              const float* __restrict__ anc, const int* __restrict__ imgsz,
              float* __restrict__ ws_score, int* __restrict__ ws_label,
              float4* __restrict__ ws_obox, int A, int BA) {
    int g = blockIdx.x * blockDim.x + threadIdx.x;
    if (g >= BA) return;
    int a = g % A;

    const float4* row = reinterpret_cast<const float4*>(cls + (size_t)g * NCLS);
    float m = -3.4e38f;
    int   ml = 0;
#pragma unroll
    for (int i = 0; i < NCLS / 4; ++i) {
        float4 v = row[i];
        if (v.x > m) { m = v.x; ml = 4 * i + 0; }
        if (v.y > m) { m = v.y; ml = 4 * i + 1; }
        if (v.z > m) { m = v.z; ml = 4 * i + 2; }
        if (v.w > m) { m = v.w; ml = 4 * i + 3; }
    }

    float size = (float)imgsz[0];
    float4 an = reinterpret_cast<const float4*>(anc)[a];
    float4 lc = reinterpret_cast<const float4*>(loc)[g];
    float4 bx = decode_box(an, lc, size);

    float koff = (float)ml * (2.0f * size);
    ws_obox[g]  = make_float4(bx.x + koff, bx.y + koff, bx.z + koff, bx.w + koff);
    ws_label[g] = ml;
    ws_score[g] = (m > SCORE_TH) ? m : -1.0f;
}

// ---------------------------------------------------------------------------
// Phase 2: greedy NMS. One 1024-thread block per image. Score vector in LDS.
// Each iteration fuses "suppress vs previous pick" with the next argmax in a
// single strided pass; wave32 shuffle + LDS tree argmax, lowest-index
// tie-break (matches jnp.argmax). Area recomputed from obox (bit-identical
// f32 expression) to avoid a second global stream in the hot loop.
// ---------------------------------------------------------------------------
__global__ __launch_bounds__(NMS_THREADS, 1)
void det_nms(const float4* __restrict__ ws_obox,
             const float* __restrict__ ws_score, int* __restrict__ ws_keep, int A) {
    extern __shared__ float sS[];                // A floats (rounded up to 16B)
    __shared__ float  pv[NMS_THREADS / 32];
    __shared__ int    pi[NMS_THREADS / 32];
    __shared__ float4 selB;
    __shared__ float  selA;
    __shared__ int    selHave;
    __shared__ int    stopIt;

    const int b   = blockIdx.x;
    const int tid = threadIdx.x;

    // ---- bulk copy of this image's score vector: global -> LDS ------------
    // First half via the Tensor Data Mover (one DMA op, TENSORcnt), second
    // half via per-lane async copies (ASYNCcnt); the two engines overlap.
    const float* src = ws_score + (size_t)b * A;
#ifdef HAVE_TDM
    const int tdmFloats = (A / 2) & ~3;          // 16B-aligned split point
#else
    const int tdmFloats = 0;
#endif

#ifdef HAVE_TDM
    if (tid == 0) {
        unsigned long long ga = (unsigned long long)(uintptr_t)src;
        unsigned int ldsb = (unsigned int)(uintptr_t)sS;
        unsigned int n = (unsigned int)tdmFloats;
        u32x4 g0;
        g0.x = 1u;                                            // count=1 (user D#)
        g0.y = ldsb;                                          // lds_addr [63:32]
        g0.z = (unsigned int)(ga & 0xffffffffull);            // global_addr lo
        g0.w = (unsigned int)((ga >> 32) & 0x1ffffffull)      // global_addr[56:32]
             | (2u << 30);                                    // type = 2 ("image")
        i32x8 g1;
        g1[0] = 0x20000;                                      // data_size=2 (4B)
        g1[1] = (int)((n & 0xffffu) << 16);                   // tensor_dim0 lo16
        g1[2] = (int)((n >> 16) & 0xffffu) | (1 << 16);       // tensor_dim0 hi16 | tensor_dim1=1
        g1[3] = (int)((n & 0xffffu) << 16);                   // tile_dim0 = n
        g1[4] = 1;                                            // tile_dim1 = 1
        g1[5] = (int)n;                                       // tensor_dim0_stride
        g1[6] = 0;
        g1[7] = 0;
        i32x4 gz = {0, 0, 0, 0};
#if defined(__clang_major__) && __clang_major__ >= 23
        i32x8 gz8 = {0, 0, 0, 0, 0, 0, 0, 0};
        __builtin_amdgcn_tensor_load_to_lds(g0, g1, gz, gz, gz8, 0);
#else
        __builtin_amdgcn_tensor_load_to_lds(g0, g1, gz, gz, 0);
#endif
        __builtin_amdgcn_s_wait_tensorcnt(0);
    }
#endif
    {
        const int startB = tdmFloats * 4;
        unsigned long long sbase = (unsigned long long)(uintptr_t)src;
        const int nchunks = (A * 4 - startB + 15) / 16;       // 16B per transfer
        for (int t = tid; t < nchunks; t += NMS_THREADS) {
            int byteoff = startB + t * 16;
            unsigned int ldsaddr = (unsigned int)(uintptr_t)((char*)sS + byteoff);
            asm volatile("global_load_async_to_lds_b128 %0, %1, %2 offset:0"
                         :
                         : "v"(ldsaddr), "v"(byteoff), "s"(sbase)
                         : "memory");
        }
        asm volatile("s_wait_asynccnt 0" ::: "memory");
    }
    if (tid == 0) { selHave = 0; stopIt = -1; }
    __syncthreads();

    const float4* ob_b = ws_obox + (size_t)b * A;

    for (int it = 0; it < MAXD; ++it) {
        const bool   have = (selHave != 0);
        const float4 sb   = selB;
        const float  sa   = selA;

        float lm = -3.4e38f;
        int   li = 0;
        for (int e = tid; e < A; e += NMS_THREADS) {
            if (e + NMS_THREADS < A)   // stage next chunk of the L2 stream
                __builtin_prefetch((const void*)(ob_b + e + NMS_THREADS), 0, 3);
            float s = sS[e];
            if (have && s >= 0.0f) {   // only alive candidates (dead are -1)
                float4 o  = ob_b[e];
                float ae  = (o.z - o.x) * (o.w - o.y);           // == ref area
                float iw  = fmaxf(fminf(sb.z, o.z) - fmaxf(sb.x, o.x), 0.0f);
                float ih  = fmaxf(fminf(sb.w, o.w) - fmaxf(sb.y, o.y), 0.0f);
                float inter = iw * ih;
                float iou = inter / (sa + ae - inter + 1e-8f);
                if (iou > IOU_TH) { s = -1.0f; sS[e] = s; }
            }
            if (s > lm) { lm = s; li = e; }    // strict > => lowest index wins
        }

        // wave32 reduction (value desc, index asc on ties)
#pragma unroll
        for (int msk = 16; msk > 0; msk >>= 1) {
            float ov = __shfl_xor(lm, msk, 32);
            int   oi = __shfl_xor(li, msk, 32);
            if (ov > lm || (ov == lm && oi < li)) { lm = ov; li = oi; }
        }
        if ((tid & 31) == 0) { pv[tid >> 5] = lm; pi[tid >> 5] = li; }
        __syncthreads();

        if (tid == 0) {
            float bm = pv[0];
            int   bi = pi[0];
#pragma unroll
            for (int w = 1; w < NMS_THREADS / 32; ++w) {
                if (pv[w] > bm || (pv[w] == bm && pi[w] < bi)) { bm = pv[w]; bi = pi[w]; }
            }
            const bool ok = (bm > 0.0f);
            ws_keep[b * MAXD + it] = ok ? bi : -1;
            if (ok) {
                float4 sbx = ob_b[bi];
                selB = sbx;
                selA = (sbx.z - sbx.x) * (sbx.w - sbx.y);        // == ref area[j]
                selHave = 1;
            } else {
                stopIt = it;
            }
        }
        __syncthreads();

        if (stopIt >= 0) {   // nothing alive: remaining keeps are all -1
            for (int r = stopIt + 1 + tid; r < MAXD; r += NMS_THREADS)
                ws_keep[b * MAXD + r] = -1;
            break;
        }
    }
}

// ---------------------------------------------------------------------------
// Phase 3: gather kept detections; re-decode boxes at kept index.
// Output layout (float): boxes [B*MAXD*4] | scores [B*MAXD] | labels [B*MAXD]
// ---------------------------------------------------------------------------
__global__ __launch_bounds__(256, 1)
void det_out(const float* __restrict__ loc, const float* __restrict__ anc,
             const int* __restrict__ imgsz,
             const float* __restrict__ ws_score, const int* __restrict__ ws_label,
             const int* __restrict__ ws_keep,
             float* __restrict__ out, int A, int B) {
    int g = blockIdx.x * blockDim.x + threadIdx.x;
    const int total = B * MAXD;
    if (g >= total) return;

    int b = g / MAXD;
    int k = ws_keep[g];
    bool valid = (k >= 0);
    int idx = valid ? k : 0;
    size_t gi = (size_t)b * A + idx;

    float size = (float)imgsz[0];
    float4 an = reinterpret_cast<const float4*>(anc)[idx];
    float4 lc = reinterpret_cast<const float4*>(loc)[gi];
    float4 bx = decode_box(an, lc, size);

    float4 res = valid ? bx : make_float4(0.0f, 0.0f, 1.0f, 1.0f);
    float* ob = out + (size_t)g * 4;
    ob[0] = res.x; ob[1] = res.y; ob[2] = res.z; ob[3] = res.w;

    out[(size_t)total * 4 + g] = valid ? ws_score[gi] : 0.0f;
    out[(size_t)total * 5 + g] = valid ? (float)ws_label[gi] : -1.0f;
}

// ---------------------------------------------------------------------------
extern "C" void kernel_launch(void* const* d_in, const int* in_sizes, int n_in,
                              void* d_out, int out_size, void* d_ws, size_t ws_size,
                              hipStream_t stream) {
    const float* cls   = (const float*)d_in[0];   // [B, A, 80]
    const float* loc   = (const float*)d_in[1];   // [B, A, 4]
    const float* anc   = (const float*)d_in[2];   // [A, 4]
    const int*   imgsz = (const int*)d_in[3];     // scalar

    const int A  = in_sizes[2] / 4;               // 49104
    const int B  = in_sizes[0] / (A * NCLS);      // 8
    const int BA = B * A;

    // workspace carve-up (hipMalloc base is 256B aligned; float4 region is
    // 16B aligned since 2*BA*4 bytes is a multiple of 16)
    float*  ws_score = (float*)d_ws;                       // BA floats
    int*    ws_label = (int*)(ws_score + BA);              // BA ints
    float4* ws_obox  = (float4*)(ws_label + BA);           // BA float4
    int*    ws_keep  = (int*)(ws_obox + BA);               // B*MAXD ints

    float* out = (float*)d_out;

    det_prep<<<(BA + 255) / 256, 256, 0, stream>>>(
        cls, loc, anc, imgsz, ws_score, ws_label, ws_obox, A, BA);

    const size_t shmem = (size_t)(((A * 4 + 15) / 16) * 16);  // ~196 KB < 320 KB WGP LDS
    det_nms<<<B, NMS_THREADS, shmem, stream>>>(
        ws_obox, ws_score, ws_keep, A);

    det_out<<<(B * MAXD + 255) / 256, 256, 0, stream>>>(
        loc, anc, imgsz, ws_score, ws_label, ws_keep, out, A, B);
}